// PhiCell_1005022347577
// MI455X (gfx1250) — compile-verified
//
#include <hip/hip_runtime.h>
#include <math.h>

// ---------------------------------------------------------------------------
// Hysteresis (backlash) operator as a clamp-composition monoid.
//   step: s' = clamp(s, x-0.5, x+0.5)
//   combine((a1,b1),(a2,b2)) = (clamp(a1,a2,b2), clamp(b1,a2,b2))  [associative]
//   identity = (-inf, +inf);  output o_t = clamp(s0, Lo_t, Hi_t)
// Three-pass hierarchical scan, CDNA5 async global<->LDS staging.
// ---------------------------------------------------------------------------

#define CHUNK   8192                       // elements per block
#define THREADS 256
#define IPT     (CHUNK / THREADS)          // 32 items per thread (ordered, contiguous)
#define GPT     (CHUNK / 4 / THREADS)      // 8 float4 groups per thread for staging
#define SCAN2   1024                       // pass-2 block width
// pad 4 floats every 64 to break LDS bank conflicts on per-thread contiguous reads
#define PADF(e)    ((e) + (((e) >> 6) << 2))
#define BUF_FLOATS (CHUNK + ((CHUNK >> 6) << 2))   // 8704 floats = 34,816 B

__device__ __forceinline__ float2 hyst_combine(float2 p, float2 e) {
    float2 r;
    r.x = fminf(fmaxf(p.x, e.x), e.y);
    r.y = fminf(fmaxf(p.y, e.x), e.y);
    return r;
}
__device__ __forceinline__ float2 hyst_ident() {
    return make_float2(-__builtin_inff(), __builtin_inff());
}
__device__ __forceinline__ unsigned ldsoff(const void* p) {
    // flat LDS aperture address: [63:32]=SHARED_BASE, [31:0]=LDS byte offset
    return (unsigned)(size_t)p;
}

// ---- CDNA5 async DMA: global <-> LDS (tracked with ASYNCcnt) ----
__device__ __forceinline__ void async_load_b128(unsigned lds, const void* gaddr) {
    asm volatile("global_load_async_to_lds_b128 %0, %1, off"
                 :: "v"(lds), "v"((unsigned long long)(size_t)gaddr)
                 : "memory");
}
__device__ __forceinline__ void async_store_b128(void* gaddr, unsigned lds) {
    asm volatile("global_store_async_from_lds_b128 %0, %1, off"
                 :: "v"((unsigned long long)(size_t)gaddr), "v"(lds)
                 : "memory");
}
__device__ __forceinline__ void wait_async0() {
    asm volatile("s_wait_asynccnt 0x0" ::: "memory");
}

// coalesced 128-bit async staging of one chunk into padded LDS
__device__ __forceinline__ void stage_chunk(const float* __restrict__ x,
                                            float* buf, long long g0, long long T,
                                            int tid, bool full) {
    if (full) {
#pragma unroll
        for (int j = 0; j < GPT; ++j) {
            int g = j * THREADS + tid;                 // float4 group within chunk
            unsigned loff = ldsoff(&buf[PADF(g * 4)]); // 16B aligned (pads at 64-bounds)
            async_load_b128(loff, x + g0 + (long long)g * 4);
        }
    } else {
        for (int j = 0; j < GPT; ++j) {
            int g = j * THREADS + tid;
            long long e = g0 + (long long)g * 4;
            unsigned loff = ldsoff(&buf[PADF(g * 4)]);
            if (e + 3 < T) {
                async_load_b128(loff, x + e);
            } else {
                for (int k = 0; k < 4; ++k)
                    buf[PADF(g * 4 + k)] = (e + k < T) ? x[e + k] : 0.0f;
            }
        }
    }
    wait_async0();
    __syncthreads();
}

// ordered inclusive Hillis-Steele over THREADS entries in LDS
__device__ __forceinline__ void block_scan(float2* scn, int tid) {
#pragma unroll
    for (int off = 1; off < THREADS; off <<= 1) {
        float2 v = scn[tid];
        float2 r = (tid >= off) ? hyst_combine(scn[tid - off], v) : v;
        __syncthreads();
        scn[tid] = r;
        __syncthreads();
    }
}

// ---------------- pass 1: per-block aggregates ----------------
__global__ __launch_bounds__(THREADS)
void phi_pass1(const float* __restrict__ x, const float* __restrict__ wp,
               float2* __restrict__ agg, long long T) {
    __shared__ float  buf[BUF_FLOATS];
    __shared__ float2 scn[THREADS];
    const int tid = threadIdx.x;
    const long long g0 = (long long)blockIdx.x * CHUNK;
    const bool full = (g0 + CHUNK <= T);
    const float w = wp[0];

    stage_chunk(x, buf, g0, T, tid, full);

    float2 a = hyst_ident();
    const int base = tid * IPT;
    if (full) {
#pragma unroll
        for (int k = 0; k < IPT; ++k) {
            float v = w * buf[PADF(base + k)];
            a = hyst_combine(a, make_float2(v - 0.5f, v + 0.5f));
        }
    } else {
        for (int k = 0; k < IPT; ++k) {
            long long gi = g0 + base + k;
            if (gi < T) {
                float v = w * buf[PADF(base + k)];
                a = hyst_combine(a, make_float2(v - 0.5f, v + 0.5f));
            }
        }
    }
    scn[tid] = a;
    __syncthreads();
    block_scan(scn, tid);                       // need last element only
    if (tid == 0) agg[blockIdx.x] = scn[THREADS - 1];
}

// ---------------- pass 2: exclusive scan of block aggregates ----------------
__global__ __launch_bounds__(SCAN2)
void phi_pass2(const float2* __restrict__ agg, float2* __restrict__ excl, int nb) {
    __shared__ float2 s[SCAN2];
    __shared__ float2 carry_s;
    const int tid = threadIdx.x;
    if (tid == 0) carry_s = hyst_ident();
    __syncthreads();
    for (int t0 = 0; t0 < nb; t0 += SCAN2) {
        int i = t0 + tid;
        s[tid] = (i < nb) ? agg[i] : hyst_ident();
        __syncthreads();
#pragma unroll
        for (int off = 1; off < SCAN2; off <<= 1) {
            float2 v = s[tid];
            float2 r = (tid >= off) ? hyst_combine(s[tid - off], v) : v;
            __syncthreads();
            s[tid] = r;
            __syncthreads();
        }
        float2 carry = carry_s;
        if (i < nb)
            excl[i] = (tid == 0) ? carry : hyst_combine(carry, s[tid - 1]);
        __syncthreads();
        if (tid == 0) carry_s = hyst_combine(carry, s[SCAN2 - 1]); // identity-padded: safe
        __syncthreads();
    }
}

// ---------------- pass 3: emit outputs ----------------
__global__ __launch_bounds__(THREADS)
void phi_pass3(const float* __restrict__ x, const float* __restrict__ wp,
               const float* __restrict__ s0p, const float2* __restrict__ excl,
               float* __restrict__ out, long long T) {
    __shared__ float  buf[BUF_FLOATS];
    __shared__ float2 scn[THREADS];
    const int tid = threadIdx.x;
    const long long g0 = (long long)blockIdx.x * CHUNK;
    const bool full = (g0 + CHUNK <= T);
    const float w  = wp[0];
    const float s0 = s0p[0];

    stage_chunk(x, buf, g0, T, tid, full);

    // thread aggregates (ordered)
    float2 a = hyst_ident();
    const int base = tid * IPT;
    if (full) {
#pragma unroll
        for (int k = 0; k < IPT; ++k) {
            float v = w * buf[PADF(base + k)];
            a = hyst_combine(a, make_float2(v - 0.5f, v + 0.5f));
        }
    } else {
        for (int k = 0; k < IPT; ++k) {
            long long gi = g0 + base + k;
            if (gi < T) {
                float v = w * buf[PADF(base + k)];
                a = hyst_combine(a, make_float2(v - 0.5f, v + 0.5f));
            }
        }
    }
    scn[tid] = a;
    __syncthreads();
    block_scan(scn, tid);

    // per-thread exclusive prefix = block_exclusive ∘ preceding thread aggregates
    float2 eb = excl[blockIdx.x];
    float2 p  = (tid == 0) ? eb : hyst_combine(eb, scn[tid - 1]);

    // emit outputs into LDS (overwrite staged inputs in place, in order)
    if (full) {
        float o = 0.0f;
#pragma unroll
        for (int k = 0; k < IPT; ++k) {
            float v = w * buf[PADF(base + k)];
            p = hyst_combine(p, make_float2(v - 0.5f, v + 0.5f));
            o = fminf(fmaxf(s0, p.x), p.y);
            buf[PADF(base + k)] = o;
        }
        if (g0 + base + IPT == T) out[T] = o;   // tuple output #2: final state
    } else {
        for (int k = 0; k < IPT; ++k) {
            long long gi = g0 + base + k;
            if (gi < T) {
                float v = w * buf[PADF(base + k)];
                p = hyst_combine(p, make_float2(v - 0.5f, v + 0.5f));
                float o = fminf(fmaxf(s0, p.x), p.y);
                buf[PADF(base + k)] = o;
                if (gi == T - 1) out[T] = o;
            }
        }
    }
    __syncthreads();

    // coalesced async store LDS -> global
    if (full) {
#pragma unroll
        for (int j = 0; j < GPT; ++j) {
            int g = j * THREADS + tid;
            unsigned loff = ldsoff(&buf[PADF(g * 4)]);
            async_store_b128(out + g0 + (long long)g * 4, loff);
        }
    } else {
        for (int j = 0; j < GPT; ++j) {
            int g = j * THREADS + tid;
            long long e = g0 + (long long)g * 4;
            unsigned loff = ldsoff(&buf[PADF(g * 4)]);
            if (e + 3 < T) {
                async_store_b128(out + e, loff);
            } else {
                for (int k = 0; k < 4; ++k)
                    if (e + k < T) out[e + k] = buf[PADF(g * 4 + k)];
            }
        }
    }
    wait_async0();
}

extern "C" void kernel_launch(void* const* d_in, const int* in_sizes, int n_in,
                              void* d_out, int out_size, void* d_ws, size_t ws_size,
                              hipStream_t stream) {
    const float* x  = (const float*)d_in[0];
    const float* w  = (const float*)d_in[1];
    const float* s0 = (const float*)d_in[2];
    float* out = (float*)d_out;                    // [T outputs] + [1 final state]
    long long T = (long long)in_sizes[0];
    int nb = (int)((T + CHUNK - 1) / CHUNK);       // 1024 for T = 8M

    float2* agg  = (float2*)d_ws;                  // nb aggregates
    float2* excl = agg + nb;                       // nb exclusive prefixes

    phi_pass1<<<nb, THREADS, 0, stream>>>(x, w, agg, T);
    phi_pass2<<<1, SCAN2, 0, stream>>>(agg, excl, nb);
    phi_pass3<<<nb, THREADS, 0, stream>>>(x, w, s0, excl, out, T);
}